// GraphAutoencoder_88871463289213
// MI455X (gfx1250) — compile-verified
//
#include <hip/hip_runtime.h>
#include <hip/hip_bf16.h>

#define NN 100000
#define NE 3200000
#define NG 64
#define IN_CH 64
#define HID_CH 128
#define LAT 32
#define BN_EPS 1e-5f

typedef float v2f __attribute__((ext_vector_type(2)));
typedef float v8f __attribute__((ext_vector_type(8)));

// ---------------- degree / normalization ----------------
__global__ void deg_kernel(const int* __restrict__ col, float* __restrict__ deg, int E) {
    int i = blockIdx.x * blockDim.x + threadIdx.x;
    if (i < E) unsafeAtomicAdd(&deg[col[i]], 1.0f);
}

__global__ void dis_kernel(float* __restrict__ deg, int n) {
    int i = blockIdx.x * blockDim.x + threadIdx.x;
    if (i < n) {
        float d = deg[i];
        deg[i] = (d > 0.0f) ? rsqrtf(fmaxf(d, 1.0f)) : 0.0f;   // in-place -> dis
    }
}

// ---------------- WMMA fp32 GEMM: C[MxN] = A[MxK] @ B[KxN] (+epilogue) ----------------
// EPI: 0 = none, 1 = bias + batchnorm + relu, 2 = bias only
template <int K, int N, int EPI>
__global__ __launch_bounds__(256)
void wmma_gemm(const float* __restrict__ A, const float* __restrict__ Bw,
               float* __restrict__ C, int M,
               const float* __restrict__ bias,
               const float* __restrict__ bng, const float* __restrict__ bnb,
               const float* __restrict__ bnm, const float* __restrict__ bnv) {
    constexpr int NT = N / 16;
    __shared__ float sB[K * N];
    for (int i = threadIdx.x; i < K * N; i += 256) sB[i] = Bw[i];
    __syncthreads();

    const int wave = threadIdx.x >> 5;
    const int lane = threadIdx.x & 31;
    const int mtile = blockIdx.x * 8 + wave;
    if (mtile * 16 >= M) return;
    const int mbase = mtile * 16;
    const int lm   = lane & 15;   // row (A) / col (B,C) within tile
    const int half = lane >> 4;   // K-pair select for A/B, +8 row offset for C/D

    v8f acc[NT];
#pragma unroll
    for (int nt = 0; nt < NT; ++nt)
#pragma unroll
        for (int j = 0; j < 8; ++j) acc[nt][j] = 0.0f;

    const float* arow = A + (size_t)(mbase + lm) * K;
    for (int k0 = 0; k0 < K; k0 += 4) {
        const int kb = k0 + 2 * half;
        const float2 av = *(const float2*)(arow + kb);   // aligned 8B load
        v2f a; a.x = av.x; a.y = av.y;
#pragma unroll
        for (int nt = 0; nt < NT; ++nt) {
            v2f b;
            b.x = sB[(size_t)kb * N + nt * 16 + lm];
            b.y = sB[(size_t)(kb + 1) * N + nt * 16 + lm];
            acc[nt] = __builtin_amdgcn_wmma_f32_16x16x4_f32(
                false, a, false, b, (short)0, acc[nt], false, false);
        }
    }

#pragma unroll
    for (int nt = 0; nt < NT; ++nt) {
        const int cc = nt * 16 + lm;
        float scale = 1.0f, shift = 0.0f;
        if (EPI == 1) {
            scale = bng[cc] * rsqrtf(bnv[cc] + BN_EPS);
            shift = bnb[cc] + (bias[cc] - bnm[cc]) * scale;
        } else if (EPI == 2) {
            shift = bias[cc];
        }
#pragma unroll
        for (int r = 0; r < 8; ++r) {
            const int rowi = mbase + r + 8 * half;
            float val = acc[nt][r];
            if (EPI == 1)      val = fmaxf(val * scale + shift, 0.0f);
            else if (EPI == 2) val = val + shift;
            C[(size_t)rowi * N + cc] = val;
        }
    }
}

// ---------------- edge scatter: agg[col] += h[row] * dis[row]*dis[col] ----------------
template <int C>
__global__ void scatter_kernel(const float* __restrict__ h, const int* __restrict__ row,
                               const int* __restrict__ col, const float* __restrict__ dis,
                               float* __restrict__ agg, int E) {
    constexpr int CV = C / 4;
    long long idx = (long long)blockIdx.x * blockDim.x + threadIdx.x;
    if (idx >= (long long)E * CV) return;
    const int e  = (int)(idx / CV);
    const int c4 = (int)(idx % CV);
    const int r = row[e];
    const int c = col[e];
    const float nrm = dis[r] * dis[c];
    const float4 v = *(const float4*)(h + (size_t)r * C + c4 * 4);
    float* dst = agg + (size_t)c * C + c4 * 4;
    unsafeAtomicAdd(dst + 0, v.x * nrm);
    unsafeAtomicAdd(dst + 1, v.y * nrm);
    unsafeAtomicAdd(dst + 2, v.z * nrm);
    unsafeAtomicAdd(dst + 3, v.w * nrm);
}

// ---------------- bias + batchnorm(eval) + relu ----------------
template <int C>
__global__ void bn_relu_kernel(const float* __restrict__ in, const float* __restrict__ bias,
                               const float* __restrict__ g, const float* __restrict__ b,
                               const float* __restrict__ m, const float* __restrict__ v,
                               float* __restrict__ out, int n) {
    int i = blockIdx.x * blockDim.x + threadIdx.x;
    if (i >= n) return;
    const int c = i & (C - 1);
    const float scale = g[c] * rsqrtf(v[c] + BN_EPS);
    const float val = (in[i] + bias[c] - m[c]) * scale + b[c];
    out[i] = fmaxf(val, 0.0f);
}

// ---------------- per-graph mean pool (batch sorted; binary-search bounds) ----------------
__global__ __launch_bounds__(256)
void pool_kernel(const float* __restrict__ z, const int* __restrict__ batch,
                 float* __restrict__ outg) {
    const int gid = blockIdx.x;
    __shared__ int sse[2];
    if (threadIdx.x < 2) {
        const int target = gid + (int)threadIdx.x;
        int lo = 0, hi = NN;
        while (lo < hi) { int mid = (lo + hi) >> 1; if (batch[mid] < target) lo = mid + 1; else hi = mid; }
        sse[threadIdx.x] = lo;
    }
    __syncthreads();
    const int start = sse[0], end = sse[1];
    const int c   = threadIdx.x & 31;
    const int sub = threadIdx.x >> 5;   // 0..7
    float sum = 0.0f;
    for (int i = start + sub; i < end; i += 8) sum += z[(size_t)i * LAT + c];
    __shared__ float red[8][32];
    red[sub][c] = sum;
    __syncthreads();
    if (threadIdx.x < 32) {
        float t = 0.0f;
#pragma unroll
        for (int s = 0; s < 8; ++s) t += red[s][c];
        const float cnt = (float)(end - start);
        outg[gid * LAT + c] = t / fmaxf(cnt, 1.0f);
    }
}

extern "C" void kernel_launch(void* const* d_in, const int* in_sizes, int n_in,
                              void* d_out, int out_size, void* d_ws, size_t ws_size,
                              hipStream_t stream) {
    const float* x     = (const float*)d_in[0];
    const int*   eidx  = (const int*)d_in[1];
    const int*   batch = (const int*)d_in[2];
    const float* w1    = (const float*)d_in[3];
    const float* b1    = (const float*)d_in[4];
    const float* w2    = (const float*)d_in[5];
    const float* b2    = (const float*)d_in[6];
    const float* bn1g  = (const float*)d_in[7];
    const float* bn1b  = (const float*)d_in[8];
    const float* bn1m  = (const float*)d_in[9];
    const float* bn1v  = (const float*)d_in[10];
    const float* bn2g  = (const float*)d_in[11];
    const float* bn2b  = (const float*)d_in[12];
    const float* bn2m  = (const float*)d_in[13];
    const float* bn2v  = (const float*)d_in[14];
    const float* fc1w  = (const float*)d_in[15];
    const float* fc1b  = (const float*)d_in[16];
    const float* bndg  = (const float*)d_in[17];
    const float* bndb  = (const float*)d_in[18];
    const float* bndm  = (const float*)d_in[19];
    const float* bndv  = (const float*)d_in[20];
    const float* fc2w  = (const float*)d_in[21];
    const float* fc2b  = (const float*)d_in[22];

    float* out = (float*)d_out;
    float* ws  = (float*)d_ws;

    float* dis  = ws;                               // N
    float* bufA = dis  + NN;                        // N*128
    float* bufB = bufA + (size_t)NN * HID_CH;       // N*128
    float* z1   = bufB + (size_t)NN * HID_CH;       // N*32
    float* z2   = z1   + (size_t)NN * LAT;          // N*32

    const int* row = eidx;
    const int* col = eidx + NE;

    hipMemsetAsync(dis,  0, (size_t)NN * sizeof(float), stream);
    hipMemsetAsync(bufB, 0, (size_t)NN * HID_CH * sizeof(float), stream);
    hipMemsetAsync(z2,   0, (size_t)NN * LAT * sizeof(float), stream);

    deg_kernel<<<(NE + 255) / 256, 256, 0, stream>>>(col, dis, NE);
    dis_kernel<<<(NN + 255) / 256, 256, 0, stream>>>(dis, NN);

    const int mblocks = (NN / 16 + 7) / 8;   // 6250 M-tiles, 8 waves/block

    // layer 1: h = x @ W1 ; agg ; relu(bn(agg + b1))
    wmma_gemm<IN_CH, HID_CH, 0><<<mblocks, 256, 0, stream>>>(
        x, w1, bufA, NN, nullptr, nullptr, nullptr, nullptr, nullptr);
    {
        long long tot = (long long)NE * (HID_CH / 4);
        scatter_kernel<HID_CH><<<(int)((tot + 255) / 256), 256, 0, stream>>>(
            bufA, row, col, dis, bufB, NE);
    }
    bn_relu_kernel<HID_CH><<<(NN * HID_CH + 255) / 256, 256, 0, stream>>>(
        bufB, b1, bn1g, bn1b, bn1m, bn1v, bufA, NN * HID_CH);

    // layer 2: z = h1 @ W2 ; agg ; relu(bn(agg + b2))
    wmma_gemm<HID_CH, LAT, 0><<<mblocks, 256, 0, stream>>>(
        bufA, w2, z1, NN, nullptr, nullptr, nullptr, nullptr, nullptr);
    {
        long long tot = (long long)NE * (LAT / 4);
        scatter_kernel<LAT><<<(int)((tot + 255) / 256), 256, 0, stream>>>(
            z1, row, col, dis, z2, NE);
    }
    bn_relu_kernel<LAT><<<(NN * LAT + 255) / 256, 256, 0, stream>>>(
        z2, b2, bn2g, bn2b, bn2m, bn2v, z1, NN * LAT);

    // decoder: d = relu(bn(z @ fc1 + fc1b)) ; x_hat = d @ fc2 + fc2b
    wmma_gemm<LAT, HID_CH, 1><<<mblocks, 256, 0, stream>>>(
        z1, fc1w, bufB, NN, fc1b, bndg, bndb, bndm, bndv);
    wmma_gemm<HID_CH, IN_CH, 2><<<mblocks, 256, 0, stream>>>(
        bufB, fc2w, out, NN, fc2b, nullptr, nullptr, nullptr, nullptr);

    // global mean pool -> z_graph appended after x_hat
    pool_kernel<<<NG, 256, 0, stream>>>(z1, batch, out + (size_t)NN * IN_CH);
}